// MultiBoxLoss_23648089932232
// MI455X (gfx1250) — compile-verified
//
#include <hip/hip_runtime.h>
#include <math.h>

// Problem constants (from reference): B=8, A=32768, O=16, C=2
#define BN 8
#define AN 32768
#define ON 16
#define THRESH 0.35f

typedef float v2f __attribute__((ext_vector_type(2)));
typedef float v8f __attribute__((ext_vector_type(8)));

#if defined(__gfx1250__) && __has_builtin(__builtin_amdgcn_wmma_f32_16x16x4_f32)
#define USE_WMMA 1
#else
#define USE_WMMA 0
#endif

struct RedOut {
  float ce_pos, npt, sl1, pairs;
  int num_neg[BN];
};

__device__ __forceinline__ float smoothl1(float d) {
  float ad = fabsf(d);
  return (ad < 1.f) ? 0.5f * d * d : ad - 0.5f;
}

// Deterministic fixed-order block tree reduction over 256 threads.
__device__ __forceinline__ float block_sum256(float v, float* red) {
  const int tid = threadIdx.x;
  __syncthreads();
  red[tid] = v;
  __syncthreads();
  for (int s = 128; s > 0; s >>= 1) {
    if (tid < s) red[tid] += red[tid + s];
    __syncthreads();
  }
  return red[0];
}

// ---------------- kernel 0: init argmax keys ----------------
__global__ void k_init(unsigned long long* keys) {
  keys[threadIdx.x] = 0ull;  // 128 keys (B*O)
}

// ---------------- kernel 1: per-(b,o) argmax anchor via packed atomicMax ----
// key = (bits(iou) << 32) | (0x7FFFFFFF - a)  -> max key == max iou, ties -> smallest a
__global__ void k_iou(const float* __restrict__ tb, const float* __restrict__ priors,
                      unsigned long long* __restrict__ keys) {
  __shared__ float tbs[ON * 4];
  __shared__ unsigned long long skey[ON];
  const int tid = threadIdx.x;
  const int b = blockIdx.y;
  const int a = blockIdx.x * 256 + tid;
  if (tid < ON * 4) tbs[tid] = tb[b * ON * 4 + tid];
  if (tid < ON) skey[tid] = 0ull;
  __syncthreads();
  float cx = priors[a*4+0], cy = priors[a*4+1], pw = priors[a*4+2], ph = priors[a*4+3];
  float x1 = cx - pw*0.5f, y1 = cy - ph*0.5f, x2 = cx + pw*0.5f, y2 = cy + ph*0.5f;
  float areaP = (x2 - x1) * (y2 - y1);
  for (int o = 0; o < ON; o++) {
    const float* t = &tbs[o*4];
    float ix1 = fmaxf(x1, t[0]), iy1 = fmaxf(y1, t[1]);
    float ix2 = fminf(x2, t[2]), iy2 = fminf(y2, t[3]);
    float iw = fmaxf(ix2 - ix1, 0.f), ih = fmaxf(iy2 - iy1, 0.f);
    float inter = iw * ih;
    float areaB = (t[2] - t[0]) * (t[3] - t[1]);
    float iou = inter / (areaP + areaB - inter);
    unsigned long long key = ((unsigned long long)__float_as_uint(iou) << 32)
                           | (unsigned long long)(0x7FFFFFFFu - (unsigned)a);
    atomicMax(&skey[o], key);
  }
  __syncthreads();
  if (tid < ON) atomicMax(&keys[b*ON + tid], skey[tid]);
}

// ---------------- kernel 2: masks + CE (WMMA einsum) + localization --------
// Block = 256 threads = 8 waves; each wave owns a 16-anchor tile and computes
// the 16x16 ce_pair tile via V_WMMA_F32_16X16X4_F32 (K = C=2, zero padded).
__global__ void k_main(const float* __restrict__ pcls, const float* __restrict__ pbox,
                       const int* __restrict__ tcls, const float* __restrict__ tb,
                       const float* __restrict__ priors,
                       const unsigned long long* __restrict__ keys,
                       float* __restrict__ ceA, float* __restrict__ nlp0,
                       float* __restrict__ part) {
  __shared__ float tbs[ON * 4];
  __shared__ int tcs[ON];
  __shared__ unsigned int abest[ON];
  __shared__ unsigned int mask_s[8][16];
  __shared__ float pri_s[8][16][4];
  __shared__ float pb_s[8][16][4];
  __shared__ float lps[8][16][2];
  __shared__ float red[256];

  const int tid = threadIdx.x;
  const int wid = tid >> 5;
  const int lane = tid & 31;
  const int b = blockIdx.y;
  const int awave = blockIdx.x * 128 + wid * 16;

  if (tid < ON * 4) tbs[tid] = tb[b * ON * 4 + tid];
  if (tid < ON) tcs[tid] = tcls[b * ON + tid];
  if (tid >= 64 && tid < 64 + ON) {
    int o = tid - 64;
    abest[o] = 0x7FFFFFFFu - (unsigned)(keys[b*ON + o] & 0xFFFFFFFFull);
  }
  __syncthreads();

  float A0 = 0.f, A1 = 0.f;  // WMMA A operand: lanes 0-15 hold K=0,1; lanes 16-31 pad = 0
  if (lane < 16) {
    const int a = awave + lane;
    const size_t ga = (size_t)b * AN + a;
    float cx = priors[a*4+0], cy = priors[a*4+1], pw = priors[a*4+2], ph = priors[a*4+3];
    pri_s[wid][lane][0] = cx; pri_s[wid][lane][1] = cy;
    pri_s[wid][lane][2] = pw; pri_s[wid][lane][3] = ph;
    float x1 = cx - pw*0.5f, y1 = cy - ph*0.5f, x2 = cx + pw*0.5f, y2 = cy + ph*0.5f;
    float areaP = (x2 - x1) * (y2 - y1);
    unsigned int m = 0u;
    for (int o = 0; o < ON; o++) {
      const float* t = &tbs[o*4];
      float ix1 = fmaxf(x1, t[0]), iy1 = fmaxf(y1, t[1]);
      float ix2 = fminf(x2, t[2]), iy2 = fminf(y2, t[3]);
      float iw = fmaxf(ix2 - ix1, 0.f), ih = fmaxf(iy2 - iy1, 0.f);
      float inter = iw * ih;
      float areaB = (t[2] - t[0]) * (t[3] - t[1]);
      float iou = inter / (areaP + areaB - inter);
      unsigned int bit = (iou > THRESH) | ((unsigned)a == abest[o]);
      m |= bit << o;
    }
    mask_s[wid][lane] = m;
    pb_s[wid][lane][0] = pbox[ga*4+0];
    pb_s[wid][lane][1] = pbox[ga*4+1];
    pb_s[wid][lane][2] = pbox[ga*4+2];
    pb_s[wid][lane][3] = pbox[ga*4+3];
    // log_softmax over C=2
    float p0 = pcls[ga*2+0], p1 = pcls[ga*2+1];
    float mx = fmaxf(p0, p1);
    float lse = mx + logf(expf(p0 - mx) + expf(p1 - mx));
    float lp0v = p0 - lse, lp1v = p1 - lse;
    lps[wid][lane][0] = lp0v; lps[wid][lane][1] = lp1v;
    A0 = lp0v; A1 = lp1v;
    nlp0[ga] = -lp0v;                       // -log_probs[...,0], all anchors
    ceA[ga]  = (m != 0u) ? 0.f : -lp0v;     // ce_anchor (zeroed at positives)
  }
  __syncthreads();

  const int n = lane & 15;
  const int tcn = tcs[n];
#if USE_WMMA
  // D[m,n] = sum_c logprob[a=m,c] * onehot(tc[n])[c]  -> ce_pair = -D
  v2f Av; Av.x = A0; Av.y = A1;
  v2f Bv;
  if (lane < 16) { Bv.x = (tcn == 0) ? 1.f : 0.f; Bv.y = (tcn == 1) ? 1.f : 0.f; }
  else           { Bv.x = 0.f; Bv.y = 0.f; }
  v8f Cz = {};
  v8f D = __builtin_amdgcn_wmma_f32_16x16x4_f32(false, Av, false, Bv,
                                                (short)0, Cz, false, false);
#endif

  float ce_t = 0.f, npt_t = 0.f, sl1_t = 0.f, pairs_t = 0.f;
  const int half = (lane < 16) ? 0 : 8;
  for (int r = 0; r < 8; r++) {
    const int m = r + half;                          // anchor row in tile
    unsigned int bit = (mask_s[wid][m] >> n) & 1u;   // posf[a, o=n]
    if (bit) {
      pairs_t += 1.f;
#if USE_WMMA
      float dval = D[r];
#else
      float dval = lps[wid][m][0] * ((tcn == 0) ? 1.f : 0.f)
                 + lps[wid][m][1] * ((tcn == 1) ? 1.f : 0.f);
#endif
      ce_t += -dval;                                  // ce_pos
      npt_t += (tcn > 0) ? 1.f : 0.f;                 // n_pos_t
      // localization: SmoothL1(pred_box - encode(gt, prior))
      const float* t = &tbs[n*4];
      float mcx = (t[0] + t[2]) * 0.5f, mcy = (t[1] + t[3]) * 0.5f;
      float mw = t[2] - t[0], mh = t[3] - t[1];
      float prcx = pri_s[wid][m][0], prcy = pri_s[wid][m][1];
      float prw  = pri_s[wid][m][2], prh  = pri_s[wid][m][3];
      float e0 = (mcx - prcx) / (0.1f * prw);
      float e1 = (mcy - prcy) / (0.1f * prh);
      float e2 = logf(mw / prw) / 0.2f;
      float e3 = logf(mh / prh) / 0.2f;
      sl1_t += smoothl1(pb_s[wid][m][0] - e0) + smoothl1(pb_s[wid][m][1] - e1)
             + smoothl1(pb_s[wid][m][2] - e2) + smoothl1(pb_s[wid][m][3] - e3);
    }
  }

  const int blockId = b * 256 + blockIdx.x;
  float s;
  s = block_sum256(ce_t, red);    if (tid == 0) part[blockId*4 + 0] = s;
  s = block_sum256(npt_t, red);   if (tid == 0) part[blockId*4 + 1] = s;
  s = block_sum256(sl1_t, red);   if (tid == 0) part[blockId*4 + 2] = s;
  s = block_sum256(pairs_t, red); if (tid == 0) part[blockId*4 + 3] = s;
}

// ---------------- kernel 3: deterministic partial reduce + num_neg --------
__global__ void k_reduce(const float* __restrict__ part, RedOut* __restrict__ ro) {
  __shared__ float r0[256], r1[256], r2[256], r3[256];
  const int tid = threadIdx.x;
  float s0 = 0.f, s1 = 0.f, s2 = 0.f, s3 = 0.f;
  for (int k = 0; k < 8; k++) {       // rows [8t, 8t+8) -> all within batch t/32
    int row = tid * 8 + k;
    s0 += part[row*4 + 0]; s1 += part[row*4 + 1];
    s2 += part[row*4 + 2]; s3 += part[row*4 + 3];
  }
  // per-batch positive-pair count: wave32 == one batch's 32 threads
  float p = s3;
  for (int off = 16; off > 0; off >>= 1) p += __shfl_down(p, off);
  if ((tid & 31) == 0) {
    int bb = tid >> 5;
    int np = (int)p;
    int nn = 7 * np;                   // NEGPOS * num_pos
    if (nn > AN - 1) nn = AN - 1;
    if (nn < 0) nn = 0;
    ro->num_neg[bb] = nn;
  }
  r0[tid] = s0; r1[tid] = s1; r2[tid] = s2; r3[tid] = s3;
  __syncthreads();
  for (int s = 128; s > 0; s >>= 1) {
    if (tid < s) { r0[tid] += r0[tid+s]; r1[tid] += r1[tid+s];
                   r2[tid] += r2[tid+s]; r3[tid] += r3[tid+s]; }
    __syncthreads();
  }
  if (tid == 0) { ro->ce_pos = r0[0]; ro->npt = r1[0]; ro->sl1 = r2[0]; ro->pairs = r3[0]; }
}

// ---------------- kernel 4: exact stable descending rank (hard negatives) --
// rank_i = #{j: ce_j > ce_i} + #{j<i: ce_j == ce_i}; negative iff rank < num_neg.
// CE vector staged in LDS in two 64KB halves; 4 anchors per thread.
__global__ void k_rank(const float* __restrict__ ceA, const float* __restrict__ nlp0,
                       const RedOut* __restrict__ ro, float* __restrict__ rpart) {
  __shared__ float sce[AN / 2];
  __shared__ float red[256];
  const int tid = threadIdx.x;
  const int b = blockIdx.y;
  const int nneg = ro->num_neg[b];
  const int idx0 = blockIdx.x * 1024 + tid * 4;
  float ci[4];
  int rank[4];
  for (int k = 0; k < 4; k++) { ci[k] = ceA[(size_t)b*AN + idx0 + k]; rank[k] = 0; }
  for (int h = 0; h < 2; h++) {
    __syncthreads();
    for (int i = tid; i < AN/2; i += 256) sce[i] = ceA[(size_t)b*AN + h*(AN/2) + i];
    __syncthreads();
    for (int j = 0; j < AN/2; j++) {
      float cj = sce[j];
      int jg = h*(AN/2) + j;
      for (int k = 0; k < 4; k++) {
        rank[k] += (int)((cj > ci[k]) || ((cj == ci[k]) && (jg < idx0 + k)));
      }
    }
  }
  float sum = 0.f;
  for (int k = 0; k < 4; k++)
    if (rank[k] < nneg) sum += nlp0[(size_t)b*AN + idx0 + k];  // -log_probs[...,0]
  float s = block_sum256(sum, red);
  if (tid == 0) rpart[b*32 + blockIdx.x] = s;
}

// ---------------- kernel 5: finalize ----------------
__global__ void k_final(const float* __restrict__ rpart, const RedOut* __restrict__ ro,
                        float* __restrict__ out) {
  if (threadIdx.x == 0) {
    float ce_neg = 0.f;
    for (int i = 0; i < 256; i++) ce_neg += rpart[i];   // fixed order
    float loss_c = 2.f * (ro->ce_pos + ce_neg) / ro->npt;
    float cnt = ro->pairs * 4.f;
    float loss_b = ro->sl1 / fmaxf(cnt, 1.f);
    out[0] = loss_c;
    out[1] = loss_b;
    out[2] = loss_c + loss_b;
  }
}

extern "C" void kernel_launch(void* const* d_in, const int* in_sizes, int n_in,
                              void* d_out, int out_size, void* d_ws, size_t ws_size,
                              hipStream_t stream) {
  (void)in_sizes; (void)n_in; (void)out_size; (void)ws_size;
  const float* pcls  = (const float*)d_in[0];  // [B,A,2]
  const float* pbox  = (const float*)d_in[1];  // [B,A,4]
  const int*   tcls  = (const int*)  d_in[2];  // [B,O]
  const float* tbox  = (const float*)d_in[3];  // [B,O,4]
  const float* prior = (const float*)d_in[4];  // [A,4]
  float* out = (float*)d_out;

  char* ws = (char*)d_ws;
  unsigned long long* keys = (unsigned long long*)(ws);                 // 1024 B
  float* ceA   = (float*)(ws + 1024);                                   // B*A*4 = 1 MB
  float* nlp0  = (float*)(ws + 1024 + (size_t)BN*AN*4);                 // 1 MB
  float* part  = (float*)(ws + 1024 + 2*(size_t)BN*AN*4);               // 2048*16 B
  RedOut* ro   = (RedOut*)(ws + 1024 + 2*(size_t)BN*AN*4 + 2048*16);    // 48 B
  float* rpart = (float*)(ws + 1024 + 2*(size_t)BN*AN*4 + 2048*16 + 64);// 1024 B

  k_init  <<<1, 128, 0, stream>>>(keys);
  k_iou   <<<dim3(AN/256, BN), 256, 0, stream>>>(tbox, prior, keys);
  k_main  <<<dim3(AN/128, BN), 256, 0, stream>>>(pcls, pbox, tcls, tbox, prior,
                                                 keys, ceA, nlp0, part);
  k_reduce<<<1, 256, 0, stream>>>(part, ro);
  k_rank  <<<dim3(32, BN), 256, 0, stream>>>(ceA, nlp0, ro, rpart);
  k_final <<<1, 64, 0, stream>>>(rpart, ro, out);
}